// VoronoiGAT_84988812853415
// MI455X (gfx1250) — compile-verified
//
#include <hip/hip_runtime.h>
#include <math.h>

#define NN   50000
#define EE   800000
#define FIN  5
#define HH   128
#define NEGS 0.2f
#define BNEPS 1e-5f
#define ET   (EE + NN)   // edges + self loops

typedef __attribute__((ext_vector_type(2))) float v2f;
typedef __attribute__((ext_vector_type(8))) float v8f;

__device__ __forceinline__ float lrelu(float v) { return v > 0.f ? v : NEGS * v; }

// order-preserving float<->uint mapping for atomicMax on floats (incl. negatives)
__device__ __forceinline__ unsigned ordf(float f) {
    unsigned u = __float_as_uint(f);
    return (u & 0x80000000u) ? ~u : (u | 0x80000000u);
}
__device__ __forceinline__ float unordf(unsigned u) {
    unsigned v = (u & 0x80000000u) ? (u & 0x7FFFFFFFu) : ~u;
    return __uint_as_float(v);
}

// ---------------------------------------------------------------- utilities
__global__ void k_zero(float* __restrict__ p, int n) {
    int i = blockIdx.x * blockDim.x + threadIdx.x;
    if (i < n) p[i] = 0.f;
}

// ------------------------------------------------- input linear: relu(x W_in^T + b_in)
__global__ void k_in_linear(const float* __restrict__ x, const float* __restrict__ W,
                            const float* __restrict__ b, float* __restrict__ out) {
    int i = blockIdx.x * blockDim.x + threadIdx.x;
    if (i >= NN * HH) return;
    int n = i >> 7, j = i & 127;
    float acc = b[j];
#pragma unroll
    for (int k = 0; k < FIN; k++) acc += x[n * FIN + k] * W[j * FIN + k];
    out[i] = fmaxf(acc, 0.f);
}

// ------------------------------------------------- WMMA f32 GEMM: out[N,ncols] = A[N,128] * W^T
// W: [ncols,128] row-major. block = (ncols/16) waves; block handles a 16-row tile of A;
// wave w computes output columns [16w, 16w+16). Optional fused bias + relu epilogue.
__global__ __launch_bounds__(256) void k_gemm_wmma(const float* __restrict__ A,
                                                   const float* __restrict__ W,
                                                   const float* __restrict__ bias,
                                                   float* __restrict__ out,
                                                   int ncols, int doRelu) {
    __shared__ float ldsA[16 * HH];
    const int tile = blockIdx.x;
    const int tid  = threadIdx.x;
    // stage 16x128 A tile into LDS (8 KB)
    for (int i = tid; i < 16 * HH; i += blockDim.x)
        ldsA[i] = A[(tile * 16 + (i >> 7)) * HH + (i & 127)];
    __syncthreads();

    const int wave  = tid >> 5;
    const int lane  = tid & 31;
    const int row   = lane & 15;
    const int khalf = (lane < 16) ? 0 : 2;       // per ISA 32-bit A/B fragment layout
    const int col   = wave * 16 + row;           // output/B column this lane touches

    v8f c = {};
#if defined(__gfx1250__) && __has_builtin(__builtin_amdgcn_wmma_f32_16x16x4_f32)
#pragma unroll
    for (int kk = 0; kk < HH / 4; kk++) {
        v2f a, b;
        // A 16x4 f32: v0 = K{0|2}, v1 = K{1|3} (lanes<16 | lanes>=16), M = lane&15
        a.x = ldsA[row * HH + kk * 4 + khalf];
        a.y = ldsA[row * HH + kk * 4 + khalf + 1];
        // B 4x16 f32 = W^T slice: B[k][col] = W[col][k], mirrored half-split layout
        b.x = W[col * HH + kk * 4 + khalf];
        b.y = W[col * HH + kk * 4 + khalf + 1];
        c = __builtin_amdgcn_wmma_f32_16x16x4_f32(false, a, false, b, (short)0, c, false, false);
    }
#else
#warning "wmma_f32_16x16x4_f32 builtin unavailable in this pass - scalar fallback"
    for (int r = 0; r < 8; r++) {
        int orow = r + ((lane < 16) ? 0 : 8);
        float acc = 0.f;
        for (int k = 0; k < HH; k++) acc += ldsA[orow * HH + k] * W[col * HH + k];
        c[r] = acc;
    }
#endif
    // C/D layout: VGPR r, lanes<16 -> M=r, lanes>=16 -> M=r+8; N = lane&15
    float bcol = bias ? bias[col] : 0.f;
#pragma unroll
    for (int r = 0; r < 8; r++) {
        int orow  = r + ((lane < 16) ? 0 : 8);
        float val = c[r] + bcol;
        if (doRelu) val = fmaxf(val, 0.f);
        out[(tile * 16 + orow) * ncols + col] = val;
    }
}

// ------------------------------------------------- per-node per-head attention coefficients
__global__ void k_attn(const float* __restrict__ h, const float* __restrict__ a_s,
                       const float* __restrict__ a_d, float* __restrict__ sArr,
                       float* __restrict__ dArr, int heads, int cdim) {
    int i = blockIdx.x * blockDim.x + threadIdx.x;
    if (i >= NN * heads) return;
    int n = i / heads, hh = i % heads;
    float ss = 0.f, dd = 0.f;
    for (int c = 0; c < cdim; c++) {
        float v = h[n * HH + hh * cdim + c];
        ss += v * a_s[hh * cdim + c];
        dd += v * a_d[hh * cdim + c];
    }
    sArr[i] = ss;
    dArr[i] = dd;
}

// ------------------------------------------------- segment max of logits over dst
__global__ void k_edge_max(const int* __restrict__ ei, const float* __restrict__ sArr,
                           const float* __restrict__ dArr, unsigned* __restrict__ mMax,
                           int heads) {
    int e = blockIdx.x * blockDim.x + threadIdx.x;
    if (e >= ET) return;
    int src = (e < EE) ? ei[e] : (e - EE);
    int dst = (e < EE) ? ei[EE + e] : (e - EE);
    for (int hh = 0; hh < heads; hh++) {
        float logit = lrelu(sArr[src * heads + hh] + dArr[dst * heads + hh]);
        atomicMax(&mMax[dst * heads + hh], ordf(logit));
    }
}

// ------------------------------------------------- segment sum of exp(logit - max)
__global__ void k_edge_denom(const int* __restrict__ ei, const float* __restrict__ sArr,
                             const float* __restrict__ dArr, const unsigned* __restrict__ mMax,
                             float* __restrict__ denom, int heads) {
    int e = blockIdx.x * blockDim.x + threadIdx.x;
    if (e >= ET) return;
    int src = (e < EE) ? ei[e] : (e - EE);
    int dst = (e < EE) ? ei[EE + e] : (e - EE);
    for (int hh = 0; hh < heads; hh++) {
        float logit = lrelu(sArr[src * heads + hh] + dArr[dst * heads + hh]);
        float ex    = __expf(logit - unordf(mMax[dst * heads + hh]));
        atomicAdd(&denom[dst * heads + hh], ex);
    }
}

// ------------------------------------------------- weighted aggregation: agg[dst] += h[src]*alpha
// one thread per (edge, channel): 32 lanes of a wave share the same head scalars
__global__ void k_edge_agg(const int* __restrict__ ei, const float* __restrict__ h,
                           const float* __restrict__ sArr, const float* __restrict__ dArr,
                           const unsigned* __restrict__ mMax, const float* __restrict__ denom,
                           float* __restrict__ agg, int heads) {
    unsigned t = blockIdx.x * blockDim.x + threadIdx.x;
    if (t >= (unsigned)ET * HH) return;
    int e  = (int)(t >> 7);
    int ch = (int)(t & 127);
    int cdim = HH / heads;
    int hh   = ch / cdim;
    int src = (e < EE) ? ei[e] : (e - EE);
    int dst = (e < EE) ? ei[EE + e] : (e - EE);
    float logit = lrelu(sArr[src * heads + hh] + dArr[dst * heads + hh]);
    float ex    = __expf(logit - unordf(mMax[dst * heads + hh]));
    float alpha = ex / (denom[dst * heads + hh] + 1e-16f);
    atomicAdd(&agg[dst * HH + ch], h[src * HH + ch] * alpha);
}

// ------------------------------------------------- BN pass 1: per-channel sum / sumsq (bias folded in)
__global__ __launch_bounds__(128) void k_bn_stats(const float* __restrict__ agg,
                                                  const float* __restrict__ bias,
                                                  float* __restrict__ bnSum,
                                                  float* __restrict__ bnSqs) {
    int c  = threadIdx.x;
    int n0 = blockIdx.x * 64;
    float bc = bias[c];
    float s = 0.f, sq = 0.f;
    for (int k = 0; k < 64; k++) {
        int n = n0 + k;
        if (n < NN) {
            float v = agg[n * HH + c] + bc;
            s  += v;
            sq += v * v;
        }
    }
    atomicAdd(&bnSum[c], s);
    atomicAdd(&bnSqs[c], sq);
}

// ------------------------------------------------- BN pass 2: normalize (+ optional relu)
__global__ void k_bn_apply(const float* __restrict__ agg, const float* __restrict__ bias,
                           const float* __restrict__ bnSum, const float* __restrict__ bnSqs,
                           const float* __restrict__ g, const float* __restrict__ be,
                           float* __restrict__ out, int doRelu) {
    int i = blockIdx.x * blockDim.x + threadIdx.x;
    if (i >= NN * HH) return;
    int c = i & 127;
    float mean = bnSum[c] * (1.f / NN);
    float var  = bnSqs[c] * (1.f / NN) - mean * mean;
    float y = (agg[i] + bias[c] - mean) * rsqrtf(var + BNEPS) * g[c] + be[c];
    if (doRelu) y = fmaxf(y, 0.f);
    out[i] = y;
}

// ------------------------------------------------- final classifier dot (N x 64 -> N)
__global__ void k_cls2(const float* __restrict__ hc, const float* __restrict__ Wc2,
                       const float* __restrict__ bc2, float* __restrict__ out) {
    int n = blockIdx.x * blockDim.x + threadIdx.x;
    if (n >= NN) return;
    float acc = bc2[0];
    for (int j = 0; j < 64; j++) acc += hc[n * 64 + j] * Wc2[j];
    out[n] = acc;
}

// ================================================================ host side
extern "C" void kernel_launch(void* const* d_in, const int* in_sizes, int n_in,
                              void* d_out, int out_size, void* d_ws, size_t ws_size,
                              hipStream_t stream) {
    const float* x    = (const float*)d_in[0];
    const int*   ei   = (const int*)d_in[1];
    const float* W_in = (const float*)d_in[2];
    const float* b_in = (const float*)d_in[3];
    const float* W[3]    = {(const float*)d_in[4],  (const float*)d_in[8],  (const float*)d_in[12]};
    const float* a_s[3]  = {(const float*)d_in[5],  (const float*)d_in[9],  (const float*)d_in[13]};
    const float* a_d[3]  = {(const float*)d_in[6],  (const float*)d_in[10], (const float*)d_in[14]};
    const float* bL[3]   = {(const float*)d_in[7],  (const float*)d_in[11], (const float*)d_in[15]};
    const float* g[3]    = {(const float*)d_in[16], (const float*)d_in[18], (const float*)d_in[20]};
    const float* be[3]   = {(const float*)d_in[17], (const float*)d_in[19], (const float*)d_in[21]};
    const float* Wc1 = (const float*)d_in[22];
    const float* bc1 = (const float*)d_in[23];
    const float* Wc2 = (const float*)d_in[24];
    const float* bc2 = (const float*)d_in[25];
    float* out = (float*)d_out;

    // workspace layout (floats)
    float*    hA    = (float*)d_ws;             // N*H   current node features
    float*    hLin  = hA   + (size_t)NN * HH;   // N*H   post-linear
    float*    hAgg  = hLin + (size_t)NN * HH;   // N*H   aggregation accumulator
    float*    sArr  = hAgg + (size_t)NN * HH;   // N*4
    float*    dArr  = sArr + (size_t)NN * 4;    // N*4
    unsigned* mMax  = (unsigned*)(dArr + (size_t)NN * 4);  // N*4
    float*    denom = (float*)mMax + (size_t)NN * 4;       // N*4
    float*    bnSum = denom + (size_t)NN * 4;   // 128
    float*    bnSqs = bnSum + HH;               // 128
    float*    hc    = bnSqs + HH;               // N*64

    const int heads[3]  = {4, 4, 1};
    const int doRelu[3] = {1, 1, 0};

    // input layer
    k_in_linear<<<(NN * HH + 255) / 256, 256, 0, stream>>>(x, W_in, b_in, hA);

    for (int l = 0; l < 3; l++) {
        const int hd = heads[l];
        // node linear on the matrix pipe (f32 WMMA), no bias/relu here
        k_gemm_wmma<<<NN / 16, 256, 0, stream>>>(hA, W[l], (const float*)nullptr, hLin, HH, 0);
        // per-node attention scalars
        k_attn<<<(NN * hd + 255) / 256, 256, 0, stream>>>(hLin, a_s[l], a_d[l], sArr, dArr, hd, HH / hd);
        // clear accumulators
        k_zero<<<(NN * hd + 255) / 256, 256, 0, stream>>>((float*)mMax, NN * hd);
        k_zero<<<(NN * hd + 255) / 256, 256, 0, stream>>>(denom, NN * hd);
        k_zero<<<(NN * HH + 255) / 256, 256, 0, stream>>>(hAgg, NN * HH);
        k_zero<<<1, 256, 0, stream>>>(bnSum, 2 * HH);
        // segment softmax + aggregation
        k_edge_max  <<<(ET + 255) / 256, 256, 0, stream>>>(ei, sArr, dArr, mMax, hd);
        k_edge_denom<<<(ET + 255) / 256, 256, 0, stream>>>(ei, sArr, dArr, mMax, denom, hd);
        k_edge_agg  <<<((unsigned)ET * HH + 255) / 256, 256, 0, stream>>>(ei, hLin, sArr, dArr,
                                                                          mMax, denom, hAgg, hd);
        // bias + batchnorm (+relu), write back into hA for next layer
        k_bn_stats<<<(NN + 63) / 64, 128, 0, stream>>>(hAgg, bL[l], bnSum, bnSqs);
        k_bn_apply<<<(NN * HH + 255) / 256, 256, 0, stream>>>(hAgg, bL[l], bnSum, bnSqs,
                                                              g[l], be[l], hA, doRelu[l]);
    }

    // classifier head: hc = relu(hA Wc1^T + bc1) on the matrix pipe (64 cols -> 4 waves/block)
    k_gemm_wmma<<<NN / 16, 128, 0, stream>>>(hA, Wc1, bc1, hc, 64, 1);
    k_cls2<<<(NN + 255) / 256, 256, 0, stream>>>(hc, Wc2, bc2, out);
}